// PfaffianState_52484500357161
// MI455X (gfx1250) — compile-verified
//
#include <hip/hip_runtime.h>
#include <hip/hip_bf16.h>

typedef float v2f __attribute__((ext_vector_type(2)));
typedef float v8f __attribute__((ext_vector_type(8)));

#if defined(__has_builtin)
#if __has_builtin(__builtin_amdgcn_s_wait_asynccnt)
#define HAVE_WAIT_ASYNC_BUILTIN 1
#endif
#endif

// Async DMA of one 16B element from global memory into LDS
// (GLOBAL_LOAD_ASYNC_TO_LDS_B128, tracked by ASYNCcnt). Inline asm is used to
// sidestep the builtin's CUDA-LangAS-qualified pointer parameters.
// vdst = LDS byte offset (low 32 bits of a generic LDS pointer),
// vaddr = 64-bit global address, saddr = off.
__device__ __forceinline__ void async_copy16(const void* gsrc, void* ldst) {
  unsigned loff = (unsigned)(unsigned long long)ldst;
  unsigned long long ga = (unsigned long long)gsrc;
  asm volatile("global_load_async_to_lds_b128 %0, %1, off"
               :: "v"(loff), "v"(ga) : "memory");
}

__device__ __forceinline__ void wait_async0() {
#ifdef HAVE_WAIT_ASYNC_BUILTIN
  __builtin_amdgcn_s_wait_asynccnt(0);
#else
  asm volatile("s_wait_asynccnt 0" ::: "memory");
#endif
}

// One workgroup (256 threads = 8 wave32) per batch element.
// LDS layout: 64KB dynamic buffer, first used as raw gathered complex128
// (double2[4096]), then reused in place as two f32 planes Mre/Mim with
// row stride 68.
__global__ __launch_bounds__(256) void PfaffianState_52484500357161_kernel(
    const int* __restrict__ y, const double* __restrict__ F,
    double* __restrict__ out) {
  extern __shared__ __align__(16) unsigned char dynbuf[];
  double2* raw = (double2*)dynbuf;           // 4096 * 16B = 64KB
  float* Mre = (float*)dynbuf;               // 64*68*4 = 17408B
  float* Mim = Mre + 64 * 68;                // +17408B (total 34816B <= 64KB)

  __shared__ float tr_[64], ti_[64], cr_[64], ci_[64];
  __shared__ float pmag[64];
  __shared__ int sy[64];
  __shared__ int s_piv;
  __shared__ double s_vre, s_vim;

  const int tid = threadIdx.x;
  const int b = blockIdx.x;

  if (tid < 64) sy[tid] = y[b * 64 + tid];
  if (tid == 0) { s_vre = 0.0; s_vim = 0.0; }
  __syncthreads();

  // ---- Async gather: raw[i][j] = F[y[i], y[j]] (16B per lane, ASYNCcnt path)
  const double2* F2 = (const double2*)F;
#pragma unroll
  for (int p = 0; p < 16; ++p) {
    int e = tid + p * 256;
    int i = e >> 6, j = e & 63;
    async_copy16(&F2[sy[i] * 128 + sy[j]], &raw[e]);
  }
  wait_async0();
  __syncthreads();

  // ---- Form skew F_occ - F_occ^T into registers, then repurpose LDS as f32 planes
  float sre[16], sim[16];
#pragma unroll
  for (int p = 0; p < 16; ++p) {
    int e = tid + p * 256;
    int i = e >> 6, j = e & 63;
    double2 a = raw[i * 64 + j];
    double2 t = raw[j * 64 + i];
    sre[p] = (float)(a.x - t.x);
    sim[p] = (float)(a.y - t.y);
  }
  __syncthreads();
#pragma unroll
  for (int p = 0; p < 16; ++p) {
    int e = tid + p * 256;
    int i = e >> 6, j = e & 63;
    Mre[i * 68 + j] = sre[p];
    Mim[i * 68 + j] = sim[p];
  }
  __syncthreads();

  const int lane = tid & 31;
  const int l = lane & 15;
  const bool hi = lane >= 16;
  const int wv = tid >> 5;

  for (int cnt = 0; cnt < 32; ++cnt) {
    const int i = 2 * cnt;
    const int r0 = i + 2;

    // ---- pivot search over column i, rows i+1..63
    if (tid < 64) {
      float re = Mre[tid * 68 + i], im = Mim[tid * 68 + i];
      pmag[tid] = (tid >= i + 1) ? (re * re + im * im) : -1.0f;
    }
    __syncthreads();
    if (tid == 0) {
      int best = i + 1;
      float bm = pmag[i + 1];
      for (int k = i + 2; k < 64; ++k)
        if (pmag[k] > bm) { bm = pmag[k]; best = k; }
      s_piv = best;
    }
    __syncthreads();
    const int prow = s_piv;

    // ---- row swap (i+1 <-> prow), cols >= i
    if (prow != i + 1 && tid < 64 && tid >= i) {
      int j = tid;
      float t0 = Mre[(i + 1) * 68 + j];
      Mre[(i + 1) * 68 + j] = Mre[prow * 68 + j];
      Mre[prow * 68 + j] = t0;
      float t1 = Mim[(i + 1) * 68 + j];
      Mim[(i + 1) * 68 + j] = Mim[prow * 68 + j];
      Mim[prow * 68 + j] = t1;
    }
    __syncthreads();
    // ---- col swap (i+1 <-> prow), rows >= i
    if (prow != i + 1 && tid < 64 && tid >= i) {
      int k = tid;
      float t0 = Mre[k * 68 + i + 1];
      Mre[k * 68 + i + 1] = Mre[k * 68 + prow];
      Mre[k * 68 + prow] = t0;
      float t1 = Mim[k * 68 + i + 1];
      Mim[k * 68 + i + 1] = Mim[k * 68 + prow];
      Mim[k * 68 + prow] = t1;
    }
    __syncthreads();

    // ---- accumulate log pfaffian term (fp64 on scalar lane)
    if (tid == 0) {
      float m01r = Mre[i * 68 + i + 1], m01i = Mim[i * 68 + i + 1];
      float m10r = Mre[(i + 1) * 68 + i], m10i = Mim[(i + 1) * 68 + i];
      if (prow != i + 1) s_vim += 3.14159265358979323846;  // log(-1) = i*pi
      if (m10r != 0.0f || m10i != 0.0f) {
        double rr = (double)m01r, ii = (double)m01i;
        s_vre += 0.5 * log(rr * rr + ii * ii);
        s_vim += atan2(ii, rr);
      } else {
        s_vre = -__builtin_inf();
        s_vim = 0.0;
      }
    }

    // ---- build t = M[i, :]/M[i,i+1], c = M[:, i+1], zero-padded below r0
    if (tid < 64) {
      int j = tid;
      float pr = Mre[i * 68 + i + 1], pi2 = Mim[i * 68 + i + 1];
      float dn = pr * pr + pi2 * pi2;
      float inv = (dn != 0.0f) ? 1.0f / dn : 0.0f;
      float mr = Mre[i * 68 + j], mi = Mim[i * 68 + j];
      bool act = (j >= r0);
      tr_[j] = act ? (mr * pr + mi * pi2) * inv : 0.0f;
      ti_[j] = act ? (mi * pr - mr * pi2) * inv : 0.0f;
      cr_[j] = act ? Mre[j * 68 + i + 1] : 0.0f;
      ci_[j] = act ? Mim[j * 68 + i + 1] : 0.0f;
    }
    __syncthreads();

    // ---- complex skew rank-2 update via V_WMMA_F32_16X16X4_F32
    // A (16x4) = [tr ti cr ci] over rows; B_re rows = [cr -ci -tr ti],
    // B_im rows = [ci cr -ti -tr] over cols. 16 tiles, 8 waves, 2 tiles/wave.
    for (int tt = wv; tt < 16; tt += 8) {
      const int R = (tt >> 2) * 16;
      const int S = (tt & 3) * 16;
      if (R + 15 < r0 || S + 15 < r0) continue;  // wave-uniform skip

      v2f A;
      A.x = hi ? cr_[R + l] : tr_[R + l];
      A.y = hi ? ci_[R + l] : ti_[R + l];
      v2f Bre, Bim;
      Bre.x = hi ? -tr_[S + l] : cr_[S + l];
      Bre.y = hi ? ti_[S + l] : -ci_[S + l];
      Bim.x = hi ? -ti_[S + l] : ci_[S + l];
      Bim.y = hi ? -tr_[S + l] : cr_[S + l];

      float* pR = &Mre[(R + (hi ? 8 : 0)) * 68 + (S + l)];
      float* pI = &Mim[(R + (hi ? 8 : 0)) * 68 + (S + l)];
      v8f Cre, Cim;
#pragma unroll
      for (int v = 0; v < 8; ++v) {
        Cre[v] = pR[v * 68];
        Cim[v] = pI[v * 68];
      }
      Cre = __builtin_amdgcn_wmma_f32_16x16x4_f32(false, A, false, Bre,
                                                  (short)0, Cre, false, false);
      Cim = __builtin_amdgcn_wmma_f32_16x16x4_f32(false, A, false, Bim,
                                                  (short)0, Cim, false, false);
#pragma unroll
      for (int v = 0; v < 8; ++v) {
        pR[v * 68] = Cre[v];
        pI[v * 68] = Cim[v];
      }
    }
    __syncthreads();
  }

  if (tid == 0) {
    out[2 * b] = s_vre;
    out[2 * b + 1] = s_vim;
  }
}

extern "C" void kernel_launch(void* const* d_in, const int* in_sizes, int n_in,
                              void* d_out, int out_size, void* d_ws,
                              size_t ws_size, hipStream_t stream) {
  (void)in_sizes; (void)n_in; (void)out_size; (void)d_ws; (void)ws_size;
  const int* y = (const int*)d_in[0];       // y: [512, 64] indices
  const double* F = (const double*)d_in[1]; // F: [128,128] complex128 (re,im)
  double* out = (double*)d_out;             // [512] complex128 -> interleaved
  PfaffianState_52484500357161_kernel<<<dim3(512), dim3(256), 65536, stream>>>(
      y, F, out);
}